// ProvenceLoss_37563783971140
// MI455X (gfx1250) — compile-verified
//
#include <hip/hip_runtime.h>
#include <hip/hip_bf16.h>
#include <math.h>

// Problem constants (from reference)
#define BB   1024
#define TT   8192
#define CC   2
#define MSEG 128

typedef __attribute__((ext_vector_type(2))) float v2f;
typedef __attribute__((ext_vector_type(8))) float v8f;

// ---------------------------------------------------------------------------
// Kernel 1: per-row segment pooling + NLL.  Bandwidth-bound streaming kernel.
// 1 block (8 wave32) per row; wave w owns segments w, w+8, ...  Lanes stride
// the segment with float2 loads (8B/lane -> 256B coalesced per wave-load).
// Boundaries are wave-uniform -> uniform control flow around the shuffles.
// Writes row_nll[b], row_cnt[b] unconditionally (deterministic, no atomics).
// ---------------------------------------------------------------------------
__global__ __launch_bounds__(256) void prune_rows_kernel(
    const float* __restrict__ logits,   // [B,T,C] f32
    const int*   __restrict__ labels,   // [B,MS]
    const int*   __restrict__ bnds,     // [B,MS,2]  (start,end), -1 = invalid
    float* __restrict__ row_nll,        // [B]
    float* __restrict__ row_cnt)        // [B]
{
    const int b    = blockIdx.x;
    const int lane = threadIdx.x & 31;
    const int wave = threadIdx.x >> 5;

    const float2* __restrict__ row =
        (const float2*)(logits + (size_t)b * TT * CC);

    float nll_acc = 0.0f;
    float cnt_acc = 0.0f;

    for (int m = wave; m < MSEG; m += 8) {
        const int base2 = ((b * MSEG) + m) * 2;
        int s = bnds[base2 + 0];
        int e = bnds[base2 + 1];
        if (s == -1) continue;                 // wave-uniform branch
        s = min(max(s, 0), TT);
        e = min(max(e, 0), TT);

        float s0 = 0.0f, s1 = 0.0f;
        for (int t = s + lane; t < e; t += 32) {
            float2 v = row[t];                 // coalesced 8B/lane
            s0 += v.x;
            s1 += v.y;
        }
        // wave32 butterfly reduction: all lanes end with the full sums
        #pragma unroll
        for (int off = 16; off > 0; off >>= 1) {
            s0 += __shfl_xor(s0, off, 32);
            s1 += __shfl_xor(s1, off, 32);
        }

        const float len = (float)(e - s);
        float p0 = 0.0f, p1 = 0.0f;
        if (len > 0.0f) {                      // reference: pooled=0 if len==0
            const float inv = 1.0f / len;
            p0 = s0 * inv;
            p1 = s1 * inv;
        }
        // 2-way log-softmax NLL (max-subtracted, stable)
        const float mx  = fmaxf(p0, p1);
        const float lse = mx + logf(expf(p0 - mx) + expf(p1 - mx));
        const int   lab = labels[b * MSEG + m];
        const float pl  = (lab == 0) ? p0 : p1;
        nll_acc += (lse - pl);
        cnt_acc += 1.0f;
    }

    __shared__ float sh[16];
    if (lane == 0) { sh[wave] = nll_acc; sh[8 + wave] = cnt_acc; }
    __syncthreads();
    if (threadIdx.x == 0) {
        float n = 0.0f, c = 0.0f;
        #pragma unroll
        for (int w = 0; w < 8; ++w) { n += sh[w]; c += sh[8 + w]; }
        row_nll[b] = n;
        row_cnt[b] = c;
    }
}

// stable softplus
__device__ __forceinline__ float softplusf(float x) {
    return fmaxf(x, 0.0f) + log1pf(expf(-fabsf(x)));
}

// ---------------------------------------------------------------------------
// Kernel 2: single wave32.  Reduces the three 1024-element vectors (BCE terms,
// row NLL partials, row counts) with V_WMMA_F32_16X16X4_F32 against an
// all-ones B matrix: D = A*1 + C accumulates exact f32 row sums.  The lane ->
// A-element packing only needs to be a bijection (sums are permutation
// invariant), so this is robust to the exact A VGPR layout.
// ---------------------------------------------------------------------------
__global__ __launch_bounds__(32) void final_combine_kernel(
    const float* __restrict__ rlog,     // [1024] ranking logits
    const float* __restrict__ rlab,     // [1024] ranking labels
    const float* __restrict__ row_nll,  // [1024]
    const float* __restrict__ row_cnt,  // [1024]
    float* __restrict__ out)            // [1] scalar
{
    const int lane = threadIdx.x;
    const int mrow = lane & 15;
    const int half = lane >> 4;

    const v2f ones = {1.0f, 1.0f};
    v8f c_bce = {};
    v8f c_nll = {};
    v8f c_cnt = {};

    for (int chunk = 0; chunk < 16; ++chunk) {
        // bijective packing of 64 chunk elements across the wave (2 per lane)
        const int i0 = chunk * 64 + mrow * 4 + half * 2;

        v2f a;
        {
            const float x0 = rlog[i0],     y0 = rlab[i0];
            const float x1 = rlog[i0 + 1], y1 = rlab[i0 + 1];
            a[0] = y0 * softplusf(-x0) + (1.0f - y0) * softplusf(x0);
            a[1] = y1 * softplusf(-x1) + (1.0f - y1) * softplusf(x1);
        }
        c_bce = __builtin_amdgcn_wmma_f32_16x16x4_f32(
            false, a, false, ones, (short)0, c_bce, false, false);

        v2f an = { row_nll[i0], row_nll[i0 + 1] };
        c_nll = __builtin_amdgcn_wmma_f32_16x16x4_f32(
            false, an, false, ones, (short)0, c_nll, false, false);

        v2f ac = { row_cnt[i0], row_cnt[i0 + 1] };
        c_cnt = __builtin_amdgcn_wmma_f32_16x16x4_f32(
            false, ac, false, ones, (short)0, c_cnt, false, false);
    }

    // C[m][n] is identical for every n; lane n holds M rows {half*8 .. half*8+7}.
    float sb = 0.0f, sn = 0.0f, sc = 0.0f;
    #pragma unroll
    for (int r = 0; r < 8; ++r) { sb += c_bce[r]; sn += c_nll[r]; sc += c_cnt[r]; }
    sb += __shfl_xor(sb, 16, 32);
    sn += __shfl_xor(sn, 16, 32);
    sc += __shfl_xor(sc, 16, 32);

    if (lane == 0) {
        const float bce   = sb * (1.0f / (float)BB);
        const float prune = (sc > 0.0f) ? (sn / fmaxf(sc, 1.0f)) : 0.0f;
        out[0] = 1.0f * bce + 0.5f * prune;   // RANK_W=1.0, PRUNE_W=0.5
    }
}

// ---------------------------------------------------------------------------
// Launch.  d_in order: ranking_logits, ranking_labels, pruning_logits,
// pruning_labels, boundaries.  d_ws: row_nll[1024] | row_cnt[1024]; every slot
// is rewritten each call, so no zero-init needed.
// ---------------------------------------------------------------------------
extern "C" void kernel_launch(void* const* d_in, const int* in_sizes, int n_in,
                              void* d_out, int out_size, void* d_ws, size_t ws_size,
                              hipStream_t stream) {
    const float* rlog = (const float*)d_in[0];
    const float* rlab = (const float*)d_in[1];
    const float* plog = (const float*)d_in[2];
    const int*   plab = (const int*)d_in[3];
    const int*   bnds = (const int*)d_in[4];

    float* row_nll = (float*)d_ws;
    float* row_cnt = row_nll + BB;

    prune_rows_kernel<<<BB, 256, 0, stream>>>(plog, plab, bnds, row_nll, row_cnt);
    final_combine_kernel<<<1, 32, 0, stream>>>(rlog, rlab, row_nll, row_cnt,
                                               (float*)d_out);
}